// v_ProjectLayer_39144331935945
// MI455X (gfx1250) — compile-verified
//
#include <hip/hip_runtime.h>
#include <stdint.h>

// complex64 inputs arrive as interleaved (re,im) float32 pairs.
typedef float v2f __attribute__((ext_vector_type(2)));
typedef float v8f __attribute__((ext_vector_type(8)));

#define NPATH   8
#define SITES   131072          // 16*16*16*32
#define G_BYTES (64 * 288)      // 64 t-groups * 4 sites * 9 complex * 8B
#define W_BYTES (64 * 512)      // 64 t-groups * 4 sites * 16 complex * 8B
#define SMEM_BYTES (G_BYTES + W_BYTES)  // 51200 B; reduction scratch reuses it

__device__ __forceinline__ void cfma(float2& a, const float2 u, const float2 v) {
    a.x = fmaf(u.x, v.x, a.x);
    a.x = fmaf(-u.y, v.y, a.x);
    a.y = fmaf(u.x, v.y, a.y);
    a.y = fmaf(u.y, v.x, a.y);
}

__global__ __launch_bounds__(256) void vproject_kernel(
    const float* __restrict__ fea,   // [1][131072][4][3] complex -> *2 floats
    const float* __restrict__ gau,   // [8][131072][3][3] complex
    const float* __restrict__ wgt,   // [8][131072][4][4] complex
    float* __restrict__ out)         // [1][4][4][4][8][4][3] complex
{
    __shared__ __align__(16) unsigned char smem[SMEM_BYTES];

    const int tid = threadIdx.x;
    const int bid = blockIdx.x;
    // coarse site (4,4,4,8), t fastest
    const int ct = bid & 7, cz = (bid >> 3) & 3, cy = (bid >> 5) & 3, cx = bid >> 7;
    // fine offset within 4x4x4x4 block, dt fastest so 4 consecutive threads hit contiguous t
    const int dt = tid & 3, dz = (tid >> 2) & 3, dy = (tid >> 4) & 3, dx = tid >> 6;
    // fine linear site: lin = x*8192 + y*512 + z*32 + t
    const int lin = (cx * 4 + dx) * 8192 + (cy * 4 + dy) * 512 + (cz * 4 + dz) * 32 + ct * 4 + dt;
    const int linbase = cx * 32768 + cy * 2048 + cz * 128 + ct * 4; // block origin (dt=0)

    // per-thread fermion spinor: 12 complex, contiguous -> vectorizes to b128 loads
    float2 fe[12];
    {
        const float2* fp = (const float2*)(fea + (size_t)lin * 24u);
        #pragma unroll
        for (int k = 0; k < 12; ++k) fe[k] = fp[k];
    }

    float2 acc[12];
    #pragma unroll
    for (int k = 0; k < 12; ++k) acc[k] = make_float2(0.f, 0.f);

    // wave-relative LDS byte address = low 32 bits of generic pointer to shared
    const uint32_t lds_g = (uint32_t)(uintptr_t)(&smem[0]);
    const uint32_t lds_w = lds_g + G_BYTES;
    const uint64_t gbase = (uint64_t)(uintptr_t)gau;
    const uint64_t wbase = (uint64_t)(uintptr_t)wgt;

    const int gsub = (tid >> 2) * 288 + (tid & 3) * 72;   // this thread's gauge in LDS
    const int wsub = (tid >> 2) * 512 + (tid & 3) * 128;  // this thread's weights in LDS

    for (int j = 0; j < NPATH; ++j) {
        // --- stage gauge: 64 groups * 288B = 1152 x 16B pieces, coalesced async copies ---
        for (int p = tid; p < 1152; p += 256) {
            const int gi  = p / 18;
            const int off = p - gi * 18;
            const int glin = linbase + (gi >> 4) * 8192 + ((gi >> 2) & 3) * 512 + (gi & 3) * 32;
            const uint64_t ga = gbase + (uint64_t)(j * SITES + glin) * 72u + (uint32_t)(off * 16);
            const uint32_t la = lds_g + (uint32_t)(gi * 288 + off * 16);
            asm volatile("global_load_async_to_lds_b128 %0, %1, off"
                         :: "v"(la), "v"(ga) : "memory");
        }
        // --- stage weights: 64 groups * 512B = 2048 x 16B pieces ---
        for (int p = tid; p < 2048; p += 256) {
            const int gi  = p >> 5;
            const int off = p & 31;
            const int glin = linbase + (gi >> 4) * 8192 + ((gi >> 2) & 3) * 512 + (gi & 3) * 32;
            const uint64_t wa = wbase + (uint64_t)(j * SITES + glin) * 128u + (uint32_t)(off * 16);
            const uint32_t la = lds_w + (uint32_t)(gi * 512 + off * 16);
            asm volatile("global_load_async_to_lds_b128 %0, %1, off"
                         :: "v"(la), "v"(wa) : "memory");
        }
        asm volatile("s_wait_asynccnt 0x0" ::: "memory");
        __syncthreads();

        // --- per-site complex math from LDS ---
        const float2* gl = (const float2*)(smem + gsub);
        const float4* wl = (const float4*)(smem + G_BYTES + wsub);

        float2 g[9];
        #pragma unroll
        for (int q = 0; q < 9; ++q) g[q] = gl[q];

        // gt[s][a] = sum_b g[a][b] * fe[s][b]
        float2 gt[12];
        #pragma unroll
        for (int s = 0; s < 4; ++s) {
            #pragma unroll
            for (int a = 0; a < 3; ++a) {
                float2 r = make_float2(0.f, 0.f);
                #pragma unroll
                for (int b = 0; b < 3; ++b) cfma(r, g[a * 3 + b], fe[s * 3 + b]);
                gt[s * 3 + a] = r;
            }
        }
        // acc[s][c] += sum_u w[s][u] * gt[u][c]
        #pragma unroll
        for (int s = 0; s < 4; ++s) {
            const float4 wa0 = wl[s * 2 + 0];
            const float4 wa1 = wl[s * 2 + 1];
            const float2 ws[4] = { make_float2(wa0.x, wa0.y), make_float2(wa0.z, wa0.w),
                                   make_float2(wa1.x, wa1.y), make_float2(wa1.z, wa1.w) };
            #pragma unroll
            for (int c = 0; c < 3; ++c) {
                #pragma unroll
                for (int u = 0; u < 4; ++u) cfma(acc[s * 3 + c], ws[u], gt[u * 3 + c]);
            }
        }
        __syncthreads();  // LDS reused by next path's staging
    }

    // ---- pooling: reduce 256 threads x 24 floats. Tree 256->32 rows in LDS, ----
    // ---- then finish 32-way sums with chained V_WMMA_F32_16X16X4_F32.        ----
    float* P = (float*)smem;  // [256][25] floats (stride 25 vs bank conflicts) = 25600B
    #pragma unroll
    for (int k = 0; k < 12; ++k) {
        P[tid * 25 + 2 * k]     = acc[k].x;
        P[tid * 25 + 2 * k + 1] = acc[k].y;
    }
    __syncthreads();
    #pragma unroll
    for (int s = 128; s >= 32; s >>= 1) {
        if (tid < s) {
            #pragma unroll
            for (int c = 0; c < 24; ++c) P[tid * 25 + c] += P[(tid + s) * 25 + c];
        }
        __syncthreads();
    }

    if (tid < 32) {  // wave 0, EXEC all ones as WMMA requires
        const int lane = tid;
        const int m  = lane & 15;          // A-matrix row this lane owns
        const int kb = (lane >> 4) << 1;   // A layout: lanes 0-15 hold K0/K1, 16-31 hold K2/K3
        v2f bones; bones.x = 1.0f; bones.y = 1.0f;   // B = ones -> D[m][n] = rowsum(A[m])
        float* o = out + (size_t)bid * 24u;
        #pragma unroll
        for (int pass = 0; pass < 2; ++pass) {
            const int comp = pass ? (16 + (m & 7)) : m;  // component carried in row m
            v8f d = {};
            #pragma unroll
            for (int i = 0; i < 8; ++i) {  // 8 chained WMMAs sum 32 values per component
                v2f a;
                a.x = P[(4 * i + kb)     * 25 + comp];
                a.y = P[(4 * i + kb + 1) * 25 + comp];
                d = __builtin_amdgcn_wmma_f32_16x16x4_f32(
                        false, a, false, bones, (short)0, d, false, false);
            }
            // D layout: lane n(0-15) VGPR r = D[r][n]; lanes 16-31 VGPR r = D[8+r][n-16]
            if (pass == 0) {
                if (lane == 0) {
                    #pragma unroll
                    for (int r = 0; r < 8; ++r) o[r] = d[r];
                }
                if (lane == 16) {
                    #pragma unroll
                    for (int r = 0; r < 8; ++r) o[8 + r] = d[r];
                }
            } else if (lane == 0) {
                #pragma unroll
                for (int r = 0; r < 8; ++r) o[16 + r] = d[r];
            }
        }
    }
}

extern "C" void kernel_launch(void* const* d_in, const int* in_sizes, int n_in,
                              void* d_out, int out_size, void* d_ws, size_t ws_size,
                              hipStream_t stream) {
    (void)in_sizes; (void)n_in; (void)out_size; (void)d_ws; (void)ws_size;
    const float* fea = (const float*)d_in[0];  // features_in  (complex64 as float pairs)
    const float* gau = (const float*)d_in[1];  // gauge_fields
    const float* wgt = (const float*)d_in[2];  // weights
    float* out = (float*)d_out;
    hipLaunchKernelGGL(vproject_kernel, dim3(512), dim3(256), 0, stream,
                       fea, gau, wgt, out);
}